// SingleHeadAttention_61426622267527
// MI455X (gfx1250) — compile-verified
//
#include <hip/hip_runtime.h>
#include <math.h>
#include <stdint.h>

// Problem constants (from reference setup_inputs)
#define B_    256
#define NNEI  31
#define G_    32          // N+1 kv groups (group 0 = self)
#define T_    64
#define H_    128
#define XS    132         // padded LDS stride for H-wide (128) tiles  -> bank-conflict free
#define AS_   68          // padded LDS stride for T-wide (64) tiles

typedef __attribute__((ext_vector_type(2))) float v2f;
typedef __attribute__((ext_vector_type(8))) float v8f;

#if __has_builtin(__builtin_amdgcn_tensor_load_to_lds) && __has_builtin(__builtin_amdgcn_s_wait_tensorcnt)
#define USE_TDM 1
#else
#define USE_TDM 0
#endif

static __device__ __forceinline__ v8f wmma_f32(v2f a, v2f b, v8f c) {
  // D = A(16x4,f32) * B(4x16,f32) + C(16x16,f32)
  return __builtin_amdgcn_wmma_f32_16x16x4_f32(false, a, false, b, (short)0, c,
                                               false, false);
}

#if USE_TDM
typedef unsigned int u32x4 __attribute__((ext_vector_type(4)));
typedef int          i32x4 __attribute__((ext_vector_type(4)));
typedef int          i32x8 __attribute__((ext_vector_type(8)));

// TDM 2-D tile load: rows x d0 fp32 elements, row stride d0_stride (elements),
// destination LDS offset lds_off (bytes) with LDS padding pad codes
// (interval code: 128 DW -> 6, 64 DW -> 5 ; amount code: 4 DW -> 3).
static __device__ __forceinline__ void tdm_load_2d(unsigned lds_off, const void* gsrc,
                                                   unsigned d0, unsigned rows,
                                                   unsigned d0_stride,
                                                   unsigned pad_i, unsigned pad_a) {
  unsigned long long ga = (unsigned long long)(size_t)gsrc;
  u32x4 g0;
  g0[0] = 1u;                                          // count=1, user descriptor
  g0[1] = lds_off;                                     // lds_addr (bytes)
  g0[2] = (unsigned)(ga & 0xffffffffu);                // global_addr[31:0]
  g0[3] = (unsigned)((ga >> 32) & 0x1ffffffu)          // global_addr[56:32]
          | (2u << 30);                                // type = 2 ("image")
  i32x8 g1;
  g1[0] = (int)((2u << 16)                             // data_size = 4 bytes
                | (1u << 20)                           // pad_enable
                | (pad_i << 22) | (pad_a << 25));      // pad interval/amount codes
  g1[1] = (int)((d0 & 0xffffu) << 16);                 // barrier_addr=0 | tensor_dim0 lo16
  g1[2] = (int)(((d0 >> 16) & 0xffffu) | ((rows & 0xffffu) << 16)); // d0 hi | tensor_dim1 lo
  g1[3] = (int)(((rows >> 16) & 0xffffu) | ((d0 & 0xffffu) << 16)); // d1 hi | tile_dim0
  g1[4] = (int)(rows & 0xffffu);                       // tile_dim1 | tile_dim2=0
  g1[5] = (int)d0_stride;                              // tensor_dim0_stride[31:0]
  g1[6] = 0;                                           // stride hi | dim1_stride lo
  g1[7] = 0;
  i32x4 z4 = {0, 0, 0, 0};
#if __clang_major__ >= 23
  i32x8 z8 = {0, 0, 0, 0, 0, 0, 0, 0};
  __builtin_amdgcn_tensor_load_to_lds(g0, g1, z4, z4, z8, 0);
#else
  __builtin_amdgcn_tensor_load_to_lds(g0, g1, z4, z4, 0);
#endif
}
#endif  // USE_TDM

// Fallback: copy a 64x128 fp32 tile (row-major, contiguous) global -> padded LDS.
static __device__ __forceinline__ void load_tile_64xH(const float* __restrict__ src,
                                                      float* __restrict__ dstLDS,
                                                      int tid) {
#pragma unroll
  for (int i = 0; i < 8; ++i) {
    int f4  = tid + (i << 8);      // 0..2047 float4 index
    int row = f4 >> 5;             // 32 float4 per row
    int c4  = f4 & 31;
    const float4 v = reinterpret_cast<const float4*>(src)[row * 32 + c4];
    *reinterpret_cast<float4*>(dstLDS + row * XS + (c4 << 2)) = v;
  }
}

// Stage one 64x128 contiguous tile into padded LDS (TDM or manual).
static __device__ __forceinline__ void stage_tile_64xH(const float* __restrict__ src,
                                                       float* __restrict__ dstLDS,
                                                       unsigned lds_off, int tid) {
#if USE_TDM
  (void)dstLDS;
  if (tid < 32) tdm_load_2d(lds_off, src, H_, T_, H_, /*128DW*/ 6, /*4DW*/ 3);
#else
  (void)lds_off;
  load_tile_64xH(src, dstLDS, tid);
#endif
}

// Wait for staged tiles, then workgroup barrier.
static __device__ __forceinline__ void stage_wait(int tid) {
#if USE_TDM
  if (tid < 32) __builtin_amdgcn_s_wait_tensorcnt(0);
#endif
  __syncthreads();
}

// One workgroup (8 waves): C[64x128] = Xs[64x128] @ W[128x128] (W from global).
// Wave w owns column tile w (16 cols) x 4 row tiles.  K loop: 32 steps of 4.
template <bool TO_LDS>
static __device__ __forceinline__ void wave_project(const float* __restrict__ Xs,
                                                    const float* __restrict__ W,
                                                    float* __restrict__ dstLDS,
                                                    float* __restrict__ dstG,
                                                    int lane, int wave) {
  v8f acc[4] = {};
  const int col  = (wave << 4) + (lane & 15);
  const int ksel = (lane >> 4) << 1;          // lanes 0-15: K+{0,1}, 16-31: K+{2,3}
  for (int kk = 0; kk < 32; ++kk) {
    const int kb = (kk << 2) + ksel;
    v2f bf; bf.x = W[kb * H_ + col]; bf.y = W[(kb + 1) * H_ + col];
#pragma unroll
    for (int mt = 0; mt < 4; ++mt) {
      const float* ap = Xs + ((mt << 4) + (lane & 15)) * XS + kb;
      v2f af; af.x = ap[0]; af.y = ap[1];
      acc[mt] = wmma_f32(af, bf, acc[mt]);
    }
  }
  const int rofs = (lane >> 4) << 3;          // lanes 16-31 hold rows +8
#pragma unroll
  for (int mt = 0; mt < 4; ++mt) {
#pragma unroll
    for (int r = 0; r < 8; ++r) {
      const int row = (mt << 4) + rofs + r;
      if (TO_LDS) dstLDS[row * XS + col] = acc[mt][r];
      else        dstG[row * H_ + col]   = acc[mt][r];
    }
  }
}

// ---------------- Kernel 0: q = node @ Wq  (staged in d_out's output slot) ----
__global__ void k_proj_q(const float* __restrict__ node,
                         const float* __restrict__ Wq,
                         float* __restrict__ q) {
  extern __shared__ float sm[];
  float* Xs = sm;
  const int b = blockIdx.x, tid = threadIdx.x;
  stage_tile_64xH(node + (size_t)b * T_ * H_, Xs, 0u, tid);
  stage_wait(tid);
  wave_project<false>(Xs, Wq, nullptr, q + (size_t)b * T_ * H_, tid & 31, tid >> 5);
}

// ------------- Kernel 1: per (b,g) raw masked logits into A region -----------
__global__ void k_scores(const float* __restrict__ node,
                         const float* __restrict__ neigh,
                         const unsigned char* __restrict__ mask, // jax bool = 1 byte
                         const float* __restrict__ Wk,
                         const float* __restrict__ q,
                         float* __restrict__ Aout) {
  extern __shared__ float sm[];
  float* Xs = sm;                 // 64 x XS
  float* Ks = sm + 64 * XS;       // 64 x XS
  float* Qs = sm + 2 * 64 * XS;   // 64 x XS
  const int g = blockIdx.x, b = blockIdx.y, tid = threadIdx.x;
  const int lane = tid & 31, wave = tid >> 5;

  const float* Xsrc = (g == 0) ? node + (size_t)b * T_ * H_
                               : neigh + ((size_t)b * NNEI + (g - 1)) * T_ * H_;
  stage_tile_64xH(Xsrc, Xs, 0u, tid);
  stage_tile_64xH(q + (size_t)b * T_ * H_, Qs, 2u * 64 * XS * sizeof(float), tid);
  stage_wait(tid);
  wave_project<true>(Xs, Wk, Ks, nullptr, lane, wave);   // K_g into LDS
  __syncthreads();

  // S[64x64] = Q[64x128] @ K_g^T ; wave owns (mt = wave&3, nt in {2*(wave>>2)+j})
  const int mt  = wave & 3;
  const int ntb = (wave >> 2) << 1;
  v8f acc[2] = {};
  const int ksel = (lane >> 4) << 1;
  const int arow = (mt << 4) + (lane & 15);
  for (int kk = 0; kk < 32; ++kk) {
    const int kb = (kk << 2) + ksel;
    v2f af; af.x = Qs[arow * XS + kb]; af.y = Qs[arow * XS + kb + 1];
#pragma unroll
    for (int j = 0; j < 2; ++j) {
      const int ncol = ((ntb + j) << 4) + (lane & 15);
      v2f bf; bf.x = Ks[ncol * XS + kb]; bf.y = Ks[ncol * XS + kb + 1]; // B[k][n]=K[n][k]
      acc[j] = wmma_f32(af, bf, acc[j]);
    }
  }
  const int rofs = (lane >> 4) << 3;
  const unsigned char* mrow = mask + (size_t)b * T_ * T_;
  const float NEG_INF = -__builtin_inff();
#pragma unroll
  for (int j = 0; j < 2; ++j) {
    const int s = ((ntb + j) << 4) + (lane & 15);
#pragma unroll
    for (int r = 0; r < 8; ++r) {
      const int t = (mt << 4) + rofs + r;
      float v = acc[j][r];
      if (mrow[t * T_ + s]) v = NEG_INF;   // jnp.where(mask, -inf, S); TEMPERATURE==1
      Aout[(((size_t)b * T_ + t) * G_ + g) * T_ + s] = v;
    }
  }
}

// ---------------- Kernel 2: row softmax over 2048 keys, in place -------------
__global__ void k_softmax(float* __restrict__ A) {
  __shared__ float red[8];
  const int row = blockIdx.x, tid = threadIdx.x;
  const int lane = tid & 31, wave = tid >> 5;
  float* p = A + (size_t)row * (G_ * T_);
  float4 v0 = reinterpret_cast<float4*>(p)[tid];
  float4 v1 = reinterpret_cast<float4*>(p)[tid + 256];
  float vals[8] = {v0.x, v0.y, v0.z, v0.w, v1.x, v1.y, v1.z, v1.w};

  float m = vals[0];
#pragma unroll
  for (int i = 1; i < 8; ++i) m = fmaxf(m, vals[i]);
#pragma unroll
  for (int o = 16; o > 0; o >>= 1) m = fmaxf(m, __shfl_xor(m, o, 32));
  if (lane == 0) red[wave] = m;
  __syncthreads();
  float M = red[0];
#pragma unroll
  for (int i = 1; i < 8; ++i) M = fmaxf(M, red[i]);
  __syncthreads();

  float s = 0.0f;
#pragma unroll
  for (int i = 0; i < 8; ++i) { vals[i] = expf(vals[i] - M); s += vals[i]; }
#pragma unroll
  for (int o = 16; o > 0; o >>= 1) s += __shfl_xor(s, o, 32);
  if (lane == 0) red[wave] = s;
  __syncthreads();
  float S = 0.0f;
#pragma unroll
  for (int i = 0; i < 8; ++i) S += red[i];
  const float inv = 1.0f / S;

  float4 o0 = make_float4(vals[0] * inv, vals[1] * inv, vals[2] * inv, vals[3] * inv);
  float4 o1 = make_float4(vals[4] * inv, vals[5] * inv, vals[6] * inv, vals[7] * inv);
  reinterpret_cast<float4*>(p)[tid]       = o0;
  reinterpret_cast<float4*>(p)[tid + 256] = o1;
}

// -------- Kernel 3: O = sum_g A_g @ (X_g @ Wv), overwrites q staging ---------
__global__ void k_out(const float* __restrict__ node,
                      const float* __restrict__ neigh,
                      const float* __restrict__ Wv,
                      const float* __restrict__ A,
                      float* __restrict__ out) {
  extern __shared__ float sm[];
  float* Xs  = sm;                 // 64 x XS
  float* Vs  = sm + 64 * XS;       // 64 x XS
  float* Asl = sm + 2 * 64 * XS;   // 64 x AS_
  const int b = blockIdx.x, tid = threadIdx.x;
  const int lane = tid & 31, wave = tid >> 5;
  v8f acc[4] = {};
  const int ksel = (lane >> 4) << 1;
  const int ncol = (wave << 4) + (lane & 15);

  for (int g = 0; g < G_; ++g) {
    __syncthreads();   // previous iteration's fragment loads done before overwrite
    const float* Xsrc = (g == 0) ? node + (size_t)b * T_ * H_
                                 : neigh + ((size_t)b * NNEI + (g - 1)) * T_ * H_;
    const float* Arows = A + (((size_t)b * T_ + 0) * G_ + g) * T_;  // 64 rows, stride G_*T_
    stage_tile_64xH(Xsrc, Xs, 0u, tid);
#if USE_TDM
    if (tid < 32)
      tdm_load_2d(2u * 64 * XS * sizeof(float), Arows, T_, T_, G_ * T_, /*64DW*/ 5, /*4DW*/ 3);
#else
    {   // load A[b, :, g, :] (64x64) into Asl
      const int t = tid >> 2, c0 = (tid & 3) << 4;
      const float4* s4 = reinterpret_cast<const float4*>(Arows + (size_t)t * G_ * T_ + c0);
      float4* dst = reinterpret_cast<float4*>(Asl + t * AS_ + c0);
      dst[0] = s4[0]; dst[1] = s4[1]; dst[2] = s4[2]; dst[3] = s4[3];
    }
#endif
    stage_wait(tid);
    wave_project<true>(Xs, Wv, Vs, nullptr, lane, wave);   // V_g into LDS
    __syncthreads();
    // O += A_g[64x64] @ V_g[64x128]; wave owns column tile `wave`, 4 row tiles
    for (int kk = 0; kk < 16; ++kk) {
      const int kb = (kk << 2) + ksel;
      v2f bf; bf.x = Vs[kb * XS + ncol]; bf.y = Vs[(kb + 1) * XS + ncol];
#pragma unroll
      for (int mtl = 0; mtl < 4; ++mtl) {
        const int arow = (mtl << 4) + (lane & 15);
        v2f af; af.x = Asl[arow * AS_ + kb]; af.y = Asl[arow * AS_ + kb + 1];
        acc[mtl] = wmma_f32(af, bf, acc[mtl]);
      }
    }
  }
  const int rofs = (lane >> 4) << 3;
  float* orow = out + (size_t)b * T_ * H_;
#pragma unroll
  for (int mtl = 0; mtl < 4; ++mtl)
#pragma unroll
    for (int r = 0; r < 8; ++r)
      orow[((mtl << 4) + rofs + r) * H_ + ncol] = acc[mtl][r];
}

extern "C" void kernel_launch(void* const* d_in, const int* in_sizes, int n_in,
                              void* d_out, int out_size, void* d_ws, size_t ws_size,
                              hipStream_t stream) {
  (void)in_sizes; (void)n_in; (void)d_ws; (void)ws_size; (void)out_size;
  const float*         node  = (const float*)d_in[0];
  const float*         neigh = (const float*)d_in[1];
  const unsigned char* mask  = (const unsigned char*)d_in[2];
  const float*         Wq    = (const float*)d_in[3];
  const float*         Wk    = (const float*)d_in[4];
  const float*         Wv    = (const float*)d_in[5];
  float* out  = (float*)d_out;
  float* Aout = out + (size_t)B_ * T_ * H_;   // A region of the tuple output

  const size_t lds_tile = (size_t)64 * XS * sizeof(float);      // 33792 B
  k_proj_q <<<B_,              256, lds_tile,                 stream>>>(node, Wq, out);
  k_scores <<<dim3(G_, B_),    256, 3 * lds_tile,             stream>>>(node, neigh, mask, Wk, out, Aout);
  k_softmax<<<B_ * T_,         256, 0,                        stream>>>(Aout);
  k_out    <<<B_,              256, 2 * lds_tile + (size_t)64 * AS_ * sizeof(float),
                                                              stream>>>(node, neigh, Wv, Aout, out);
}